// VMFVAEWrapper_7258494730890
// MI455X (gfx1250) — compile-verified
//
#include <hip/hip_runtime.h>
#include <stdint.h>

#define B_ROWS 16384
#define DIM    2048
#define KAPPA  100.0f
#define ROWS_PER_BLOCK 8
#define NBLOCKS (B_ROWS / ROWS_PER_BLOCK)

// TDM descriptor SGPR-group vector types (clang-23 6-arg builtin form)
typedef unsigned int u32x4 __attribute__((ext_vector_type(4)));
typedef int          i32x8 __attribute__((ext_vector_type(8)));
typedef int          i32x4 __attribute__((ext_vector_type(4)));

__device__ __forceinline__ uint32_t pcg_hash(uint32_t x) {
  x = x * 747796405u + 2891336453u;
  uint32_t w = ((x >> ((x >> 28u) + 4u)) ^ x) * 277803737u;
  return (w >> 22u) ^ w;
}

__device__ __forceinline__ float u01(uint32_t h) {
  // (0,1), never 0 -> safe for log()
  return ((float)(h >> 8) + 0.5f) * (1.0f / 16777216.0f);
}

__device__ __forceinline__ void boxmuller(uint32_t h1, uint32_t h2,
                                          float* n0, float* n1) {
  float u1 = u01(h1);
  float u2 = u01(h2);
  float r = __fsqrt_rn(-2.0f * __logf(u1));
  float s, c;
  __sincosf(6.28318530718f * u2, &s, &c);
  *n0 = r * c;
  *n1 = r * s;
}

// Issue one TDM row load: 2048 f32 from gaddr -> LDS offset laddr (async).
__device__ __forceinline__ void tdm_load_row(uint64_t gaddr, uint32_t laddr) {
  u32x4 g0;
  i32x8 g1;
  i32x4 g2 = {0, 0, 0, 0};
  i32x4 g3 = {0, 0, 0, 0};
  i32x8 g4 = {0, 0, 0, 0, 0, 0, 0, 0};

  // D# group 0: [1:0]=count=1, [63:32]=lds_addr, [120:64]=global_addr, [127:126]=type=2
  g0[0] = 1u;
  g0[1] = laddr;
  g0[2] = (uint32_t)(gaddr & 0xFFFFFFFFu);
  g0[3] = (uint32_t)((gaddr >> 32) & 0x01FFFFFFu) | (2u << 30);

  // D# group 1: data_size=2 (4B); tensor_dim0=2048; tensor_dim1=1;
  //             tile_dim0=2048; tile_dim1=1; tensor_dim0_stride=2048
  g1[0] = (int)(2u << 16);                                 // data_size=4B
  g1[1] = (int)(((uint32_t)DIM & 0xFFFFu) << 16);          // tensor_dim0 lo16
  g1[2] = (int)(((uint32_t)DIM >> 16) | (1u << 16));       // dim0 hi16 | dim1 lo16
  g1[3] = (int)((uint32_t)DIM << 16);                      // tile_dim0 = 2048
  g1[4] = 1;                                               // tile_dim1 = 1
  g1[5] = DIM;                                             // dim0_stride lo32
  g1[6] = 0;
  g1[7] = 0;

  __builtin_amdgcn_tensor_load_to_lds(g0, g1, g2, g3, g4, 0);
}

// ---------------------------------------------------------------------------
// Kernel 1: per-row vMF radial weight w via Wood (1994) rejection sampling.
// Pure ALU, independent of the input tensor. One thread per row.
// ---------------------------------------------------------------------------
__global__ void vmf_w_kernel(float* __restrict__ w_out) {
  int row = blockIdx.x * blockDim.x + threadIdx.x;
  if (row >= B_ROWS) return;

  const float d  = (float)(DIM - 1);                                   // 2047
  const float b  = d / (__fsqrt_rn(4.f * KAPPA * KAPPA + d * d) + 2.f * KAPPA);
  const float x0 = (1.f - b) / (1.f + b);
  const float c  = KAPPA * x0 + d * __logf(1.f - x0 * x0);
  // Beta(a,a), a = d/2: var = 1/(4(2a+1)) -> normal approximation
  const float beta_std = __frsqrt_rn(4.f * (d + 1.f));

  uint32_t st = 0x9E3779B9u ^ (uint32_t)row * 0x9E3779B1u;
  float w = 0.f;
  bool acc = false;
#pragma unroll 1
  for (int it = 0; it < 32; ++it) {
    uint32_t h1 = pcg_hash(st + (uint32_t)(3 * it) + 1u);
    uint32_t h2 = pcg_hash(st + (uint32_t)(3 * it) + 2u);
    uint32_t h3 = pcg_hash(st + (uint32_t)(3 * it) + 3u);
    float n0, n1;
    boxmuller(h1, h2, &n0, &n1);
    float z = 0.5f + n0 * beta_std;
    z = fminf(fmaxf(z, 1e-6f), 1.f - 1e-6f);
    float wc = (1.f - (1.f + b) * z) / (1.f - (1.f - b) * z);
    float u  = u01(h3);
    bool ok = (KAPPA * wc + d * __logf(1.f - x0 * wc) - c) >= __logf(u);
    if (!acc && ok) { w = wc; acc = true; }
  }
  w_out[row] = w;
}

// ---------------------------------------------------------------------------
// Kernel 2: persistent workgroup (256 threads, 8 wave32) over 8 rows with
// double-buffered TDM:
//  - wave0 keeps up to 2 tensor_load_to_lds descriptors in flight (TENSORcnt)
//  - all threads generate Gaussian v-values while the DMA runs
//  - 3 row reductions (Σμμ, Σvv, Σμv) via wave32 shfl_xor + LDS combine
//  - output = v*A + μ*B, two FMAs/element, B128 stores
// ---------------------------------------------------------------------------
__global__ void __launch_bounds__(256) vmf_sample_kernel(
    const float* __restrict__ mu, const float* __restrict__ w_arr,
    float* __restrict__ out) {
  __shared__ __align__(16) float mu_s[2][DIM];   // ping-pong row buffers (16KB)
  __shared__ float red[8][4];

  const int tid  = threadIdx.x;
  const int base = tid * 8;
  const int row0 = blockIdx.x * ROWS_PER_BLOCK;
  const int wave = tid >> 5;
  const int lane = tid & 31;

  const uint32_t lds0 = (uint32_t)(uintptr_t)(void*)&mu_s[0][0];
  const uint32_t lds1 = (uint32_t)(uintptr_t)(void*)&mu_s[1][0];

  // Prime the pipeline: row0 -> buffer 0
  if (tid < 32) {
    tdm_load_row((uint64_t)(uintptr_t)(const void*)(mu + (size_t)row0 * DIM), lds0);
  }

#pragma unroll 1
  for (int i = 0; i < ROWS_PER_BLOCK; ++i) {
    const int row = row0 + i;
    const int buf = i & 1;

    // Prefetch next row into the other buffer (its last readers passed the
    // end-of-iteration barrier of iteration i-1, so reuse is safe).
    if (tid < 32 && (i + 1) < ROWS_PER_BLOCK) {
      tdm_load_row((uint64_t)(uintptr_t)(const void*)(mu + (size_t)(row + 1) * DIM),
                   buf ? lds0 : lds1);
    }

    // ---- Generate v (4 Box-Muller pairs) while the DMA is in flight ----
    float v[8];
    uint32_t seed = (uint32_t)row * 0x01000193u + (uint32_t)base;
#pragma unroll
    for (int p = 0; p < 4; ++p) {
      uint32_t h1 = pcg_hash(seed + (uint32_t)(2 * p));
      uint32_t h2 = pcg_hash((seed + (uint32_t)(2 * p) + 1u) ^ 0x85EBCA6Bu);
      boxmuller(h1, h2, &v[2 * p], &v[2 * p + 1]);
    }

    // Wait for the OLDER load (current row); the newer prefetch stays in flight.
    if (tid < 32) {
      if ((i + 1) < ROWS_PER_BLOCK) __builtin_amdgcn_s_wait_tensorcnt(1);
      else                          __builtin_amdgcn_s_wait_tensorcnt(0);
    }
    __syncthreads();

    // ---- Read this thread's 8 mu values from LDS (two b128 reads) ----
    float m[8];
    float4 m0 = *(const float4*)&mu_s[buf][base];
    float4 m1 = *(const float4*)&mu_s[buf][base + 4];
    m[0] = m0.x; m[1] = m0.y; m[2] = m0.z; m[3] = m0.w;
    m[4] = m1.x; m[5] = m1.y; m[6] = m1.z; m[7] = m1.w;

    float s_mm = 0.f, s_vv = 0.f, s_mv = 0.f;
#pragma unroll
    for (int k = 0; k < 8; ++k) {
      s_mm = fmaf(m[k], m[k], s_mm);
      s_vv = fmaf(v[k], v[k], s_vv);
      s_mv = fmaf(m[k], v[k], s_mv);
    }

    // wave32 butterfly reduction
#pragma unroll
    for (int off = 16; off > 0; off >>= 1) {
      s_mm += __shfl_xor(s_mm, off, 32);
      s_vv += __shfl_xor(s_vv, off, 32);
      s_mv += __shfl_xor(s_mv, off, 32);
    }
    if (lane == 0) {
      red[wave][0] = s_mm;
      red[wave][1] = s_vv;
      red[wave][2] = s_mv;
    }
    __syncthreads();

    float t_mm = 0.f, t_vv = 0.f, t_mv = 0.f;
#pragma unroll
    for (int wv = 0; wv < 8; ++wv) {
      t_mm += red[wv][0];
      t_vv += red[wv][1];
      t_mv += red[wv][2];
    }

    // ---- Closed-form combine:
    // ortho = v - mu*(t_mv/t_mm); ||ortho||^2 = t_vv - t_mv^2/t_mm
    // out = ortho/||ortho|| * sqrt(1-w^2) + mu * w/||mu||
    //     = v*A + mu*(w/||mu|| - cproj*A)
    const float w      = w_arr[row];
    const float inv_mm = __frcp_rn(t_mm);
    const float cproj  = t_mv * inv_mm;
    float s_oo = fmaxf(t_vv - t_mv * cproj, 1e-20f);
    const float A  = __fsqrt_rn(fmaxf(1.f - w * w, 0.f)) * __frsqrt_rn(s_oo);
    const float Bc = w * __frsqrt_rn(t_mm) - cproj * A;

    float4 o0, o1;
    o0.x = fmaf(m[0], Bc, v[0] * A);
    o0.y = fmaf(m[1], Bc, v[1] * A);
    o0.z = fmaf(m[2], Bc, v[2] * A);
    o0.w = fmaf(m[3], Bc, v[3] * A);
    o1.x = fmaf(m[4], Bc, v[4] * A);
    o1.y = fmaf(m[5], Bc, v[5] * A);
    o1.z = fmaf(m[6], Bc, v[6] * A);
    o1.w = fmaf(m[7], Bc, v[7] * A);

    float* orow = out + (size_t)row * DIM + base;
    *(float4*)&orow[0] = o0;
    *(float4*)&orow[4] = o1;

    // All readers done with mu_s[buf] before iteration i+1 prefetches into
    // the opposite buffer (and before i+2 reuses this one).
    __syncthreads();
  }

  // kl penalty output (single zero appended after the sampled tensor)
  if (blockIdx.x == 0 && tid == 0) out[(size_t)B_ROWS * DIM] = 0.f;
}

extern "C" void kernel_launch(void* const* d_in, const int* in_sizes, int n_in,
                              void* d_out, int out_size, void* d_ws, size_t ws_size,
                              hipStream_t stream) {
  (void)in_sizes; (void)n_in; (void)out_size; (void)ws_size;
  const float* mu  = (const float*)d_in[0];
  float*       out = (float*)d_out;
  float*       w_ws = (float*)d_ws;  // B_ROWS floats of scratch

  vmf_w_kernel<<<B_ROWS / 256, 256, 0, stream>>>(w_ws);
  vmf_sample_kernel<<<NBLOCKS, 256, 0, stream>>>(mu, w_ws, out);
}